// AdaptiveGCNdiff_63462436765701
// MI455X (gfx1250) — compile-verified
//
#include <hip/hip_runtime.h>
#include <hip/hip_bf16.h>
#include <math.h>

// ---------------------------------------------------------------------------
// Sizes (compile-time constants from the reference)
// ---------------------------------------------------------------------------
#define B_SZ   8192
#define N_SZ   17
#define D_SZ   96
#define L_SZ   5
#define H_SZ   4
#define DK_SZ  24
#define EMD_SZ 384
#define GH_SZ  128

typedef float v2f __attribute__((ext_vector_type(2)));
typedef float v8f __attribute__((ext_vector_type(8)));

// ---------------------------------------------------------------------------
// WMMA GEMM:  Y[M,CO] = epilogue( X[M,KDIM] @ W[KDIM,CO] + bias )
//   one wave -> MT vertically-stacked 16x16 output tiles (64 rows for MT=4),
//   K stepped by 4 with v_wmma_f32_16x16x4_f32 (fp32 matrix pipe -> fp32-exact
//   vs the reference, just re-associated accumulation).
// B fragment is shared across the MT tiles -> 1.5 vmem ops per WMMA, and with
// CO/KDIM compile-time all loads fold to immediate offsets off one base.
// A frag (16x4 f32): lanes 0-15 hold row m0+l, k=kb+0..1; lanes 16-31 k=kb+2..3
// B frag (4x16 f32): lanes 0-15 hold col n0+l, k=kb+0..1; lanes 16-31 k=kb+2..3
// C/D (16x16 f32):   vgpr r -> row r + 8*(lane>=16), col = n0 + (lane&15)
// ACT: 0=none 1=relu 2=swish ; RESID: Y = R + act(...) ; EXTRA: += extra[b,:]
// ---------------------------------------------------------------------------
template<int KDIM, int CO, int MT, int ACT, bool RESID, bool EXTRA>
__global__ __launch_bounds__(256)
void gemm_wmma(const float* __restrict__ X, const float* __restrict__ W,
               const float* __restrict__ bias, const float* __restrict__ R,
               const float* __restrict__ extra, float* __restrict__ Y,
               int M, int nPerB)
{
    const int wave = blockIdx.x * (blockDim.x >> 5) + (threadIdx.x >> 5);
    const int lane = threadIdx.x & 31;
    constexpr int NTN = CO >> 4;
    const int gm   = wave / NTN;              // group of MT m-tiles
    const int nt   = wave - gm * NTN;
    const int m0   = gm * (16 * MT);
    if (m0 >= M) return;
    const int n0   = nt << 4;
    const int half = lane >> 4;   // 0: k 0..1 | 1: k 2..3 (and rows +8 for C/D)
    const int l16  = lane & 15;

    const float* arow = X + (size_t)(m0 + l16) * KDIM + (half << 1);
    const float* bcol = W + (size_t)(half << 1) * CO + (n0 + l16);

    v8f acc[MT];
#pragma unroll
    for (int mi = 0; mi < MT; ++mi)
        acc[mi] = (v8f){0.f, 0.f, 0.f, 0.f, 0.f, 0.f, 0.f, 0.f};

#pragma unroll
    for (int kb = 0; kb < KDIM; kb += 4) {
        v2f b;
        b.x = bcol[kb * CO];                  // immediate offsets (CO constexpr)
        b.y = bcol[kb * CO + CO];
#pragma unroll
        for (int mi = 0; mi < MT; ++mi) {
            v2f a = *(const v2f*)(arow + (size_t)mi * 16 * KDIM + kb);
            acc[mi] = __builtin_amdgcn_wmma_f32_16x16x4_f32(
                          false, a, false, b, (short)0, acc[mi], false, false);
        }
    }

    const float bv = bias[n0 + l16];
#pragma unroll
    for (int mi = 0; mi < MT; ++mi) {
#pragma unroll
        for (int r = 0; r < 8; ++r) {
            int   m = m0 + mi * 16 + r + (half << 3);
            float t = acc[mi][r] + bv;
            if (ACT == 1) t = fmaxf(t, 0.f);
            if (ACT == 2) t = t / (1.f + expf(-t));          // swish
            if (EXTRA)    t += extra[(size_t)(m / nPerB) * CO + (n0 + l16)];
            size_t oi = (size_t)m * CO + (n0 + l16);
            if (RESID)    t += R[oi];
            Y[oi] = t;
        }
    }
}

// ---------------------------------------------------------------------------
// Graph mix: Z[row, 0:K] = X[row, :], Z[row, K:2K] = (G @ X_b)[n, :]
//   row = b*17 + n, G is a 17x17 normalized Laplacian.
// This turns every ChebConv into a single contiguous GEMM with W = [W0;W1].
// ---------------------------------------------------------------------------
__global__ __launch_bounds__(256)
void graphmix_kernel(const float* __restrict__ X, const float* __restrict__ G,
                     float* __restrict__ Z, int K, int total)
{
    int idx = blockIdx.x * 256 + threadIdx.x;
    if (idx >= total) return;
    int k   = idx % K;
    int row = idx / K;
    int n   = row % N_SZ;
    size_t base = (size_t)(row - n) * K;        // start of batch b
    const float* g = G + n * N_SZ;
    float s = 0.f;
#pragma unroll
    for (int m = 0; m < N_SZ; ++m) s += g[m] * X[base + (size_t)m * K + k];
    float* zr = Z + (size_t)row * (2 * K);
    zr[k]     = X[(size_t)row * K + k];
    zr[K + k] = s;
}

// ---------------------------------------------------------------------------
// Normalized Laplacian: L = I - D^-1/2 A D^-1/2 (adj + 5 learnable a_hat)
// ---------------------------------------------------------------------------
__global__ void lap_kernel(const float* __restrict__ adj,
                           const float* __restrict__ a_hat,
                           float* __restrict__ laps)
{
    int mat = blockIdx.x;                       // 0 = adj, 1..5 = a_hat[l]
    const float* A = (mat == 0) ? adj : (a_hat + (size_t)(mat - 1) * N_SZ * N_SZ);
    __shared__ float sA[N_SZ * N_SZ];
    __shared__ float dinv[N_SZ];
    int tid = threadIdx.x;                      // 0..288
    sA[tid] = A[tid];
    __syncthreads();
    int n = tid / N_SZ, m = tid % N_SZ;
    if (m == 0) {
        float s = 0.f;
        for (int j = 0; j < N_SZ; ++j) s += sA[n * N_SZ + j];
        dinv[n] = 1.f / sqrtf(s);
    }
    __syncthreads();
    float v = ((n == m) ? 1.f : 0.f) - dinv[n] * sA[tid] * dinv[m];
    laps[(size_t)mat * N_SZ * N_SZ + tid] = v;
}

// ---------------------------------------------------------------------------
// Timestep embedding: [sin(t*f), cos(t*f)], half=48, f_j = exp(-ln(1e4)/47 * j)
// ---------------------------------------------------------------------------
__global__ void temb_emb_kernel(const int* __restrict__ t, float* __restrict__ E,
                                int total)
{
    int idx = blockIdx.x * 256 + threadIdx.x;
    if (idx >= total) return;
    int d = idx % D_SZ;
    int b = idx / D_SZ;
    int j = (d < 48) ? d : d - 48;
    float freq = expf(-0.19596468876545072f * (float)j);   // -ln(10000)/47
    float arg  = (float)t[b] * freq;
    E[idx] = (d < 48) ? sinf(arg) : cosf(arg);
}

// ---------------------------------------------------------------------------
// GraFormer LayerNorm (torch unbiased std): wave32 per row of 96
// ---------------------------------------------------------------------------
__global__ __launch_bounds__(256)
void lnorm_kernel(const float* __restrict__ X, const float* __restrict__ ga,
                  const float* __restrict__ gb, float* __restrict__ Y, int rows)
{
    int wave = blockIdx.x * (blockDim.x >> 5) + (threadIdx.x >> 5);
    int lane = threadIdx.x & 31;
    if (wave >= rows) return;
    const float* xp = X + (size_t)wave * D_SZ;
    float v0 = xp[lane], v1 = xp[lane + 32], v2 = xp[lane + 64];
    float s = v0 + v1 + v2;
#pragma unroll
    for (int off = 16; off > 0; off >>= 1) s += __shfl_xor(s, off, 32);
    float mu = s * (1.f / 96.f);
    float d0 = v0 - mu, d1 = v1 - mu, d2 = v2 - mu;
    float q = d0 * d0 + d1 * d1 + d2 * d2;
#pragma unroll
    for (int off = 16; off > 0; off >>= 1) q += __shfl_xor(q, off, 32);
    // std = sqrt(var * n/(n-1)) = sqrt(q/95)
    float inv = 1.f / (sqrtf(q * (1.f / 95.f)) + 1e-6f);
    float* yp = Y + (size_t)wave * D_SZ;
    yp[lane]      = ga[lane]      * (d0 * inv) + gb[lane];
    yp[lane + 32] = ga[lane + 32] * (d1 * inv) + gb[lane + 32];
    yp[lane + 64] = ga[lane + 64] * (d2 * inv) + gb[lane + 64];
}

// ---------------------------------------------------------------------------
// Attention core: one thread per (b, h, n): 17x17 scores, softmax, @V (dk=24)
// ---------------------------------------------------------------------------
__global__ __launch_bounds__(256)
void attn_kernel(const float* __restrict__ Q, const float* __restrict__ Km,
                 const float* __restrict__ V, const int* __restrict__ mask,
                 float* __restrict__ O, int total)
{
    int idx = blockIdx.x * 256 + threadIdx.x;
    if (idx >= total) return;
    int n  = idx % N_SZ;
    int tt = idx / N_SZ;
    int h  = tt & (H_SZ - 1);
    int b  = tt >> 2;
    const size_t rowq = ((size_t)b * N_SZ + n) * D_SZ + h * DK_SZ;
    float qr[DK_SZ];
#pragma unroll
    for (int d = 0; d < DK_SZ; ++d) qr[d] = Q[rowq + d];
    float sc[N_SZ];
    float mx = -1e30f;
#pragma unroll
    for (int m = 0; m < N_SZ; ++m) {
        const float* kp = Km + ((size_t)b * N_SZ + m) * D_SZ + h * DK_SZ;
        float s = 0.f;
#pragma unroll
        for (int d = 0; d < DK_SZ; ++d) s += qr[d] * kp[d];
        s *= 0.20412414523193154f;                 // 1/sqrt(24)
        if (mask[b * N_SZ + m] == 0) s = -1e9f;
        sc[m] = s;
        mx = fmaxf(mx, s);
    }
    float sum = 0.f;
#pragma unroll
    for (int m = 0; m < N_SZ; ++m) { sc[m] = expf(sc[m] - mx); sum += sc[m]; }
    float inv = 1.f / sum;
    float o[DK_SZ];
#pragma unroll
    for (int d = 0; d < DK_SZ; ++d) o[d] = 0.f;
#pragma unroll
    for (int m = 0; m < N_SZ; ++m) {
        float w = sc[m] * inv;
        const float* vp = V + ((size_t)b * N_SZ + m) * D_SZ + h * DK_SZ;
#pragma unroll
        for (int d = 0; d < DK_SZ; ++d) o[d] += w * vp[d];
    }
#pragma unroll
    for (int d = 0; d < DK_SZ; ++d) O[rowq + d] = o[d];
}

// ---------------------------------------------------------------------------
// Input ChebConv (K=5 -> 96), done in two tiny VALU kernels
// ---------------------------------------------------------------------------
__global__ void xl_kernel(const float* __restrict__ x, const float* __restrict__ Lap,
                          float* __restrict__ XL, int total)
{
    int idx = blockIdx.x * 256 + threadIdx.x;
    if (idx >= total) return;
    int i   = idx % 5;
    int row = idx / 5;
    int n   = row % N_SZ;
    size_t base = (size_t)(row - n) * 5;
    const float* g = Lap + n * N_SZ;
    float s = 0.f;
#pragma unroll
    for (int m = 0; m < N_SZ; ++m) s += g[m] * x[base + (size_t)m * 5 + i];
    XL[idx] = s;
}

__global__ void incheb_kernel(const float* __restrict__ x, const float* __restrict__ XL,
                              const float* __restrict__ w, const float* __restrict__ bias,
                              float* __restrict__ OUT, int total)
{
    int idx = blockIdx.x * 256 + threadIdx.x;
    if (idx >= total) return;
    int d   = idx % D_SZ;
    int row = idx / D_SZ;
    float acc = bias[d];
#pragma unroll
    for (int i = 0; i < 5; ++i)
        acc += x[(size_t)row * 5 + i]  * w[i * D_SZ + d]
             + XL[(size_t)row * 5 + i] * w[5 * D_SZ + i * D_SZ + d];
    OUT[idx] = acc;
}

// ---------------------------------------------------------------------------
// Output ChebConv: Z[BN,192] @ w_out[192,3] + b_out
// ---------------------------------------------------------------------------
__global__ void outcheb_kernel(const float* __restrict__ Z, const float* __restrict__ w,
                               const float* __restrict__ bias, float* __restrict__ out,
                               int total)
{
    int idx = blockIdx.x * 256 + threadIdx.x;
    if (idx >= total) return;
    int c   = idx % 3;
    int row = idx / 3;
    const float* z = Z + (size_t)row * 192;
    float acc = bias[c];
#pragma unroll 8
    for (int k = 0; k < 192; ++k) acc += z[k] * w[k * 3 + c];
    out[idx] = acc;
}

// ---------------------------------------------------------------------------
// Host orchestration
// ---------------------------------------------------------------------------
static inline int gemm_blocks(size_t M, int Co, int MT)
{
    size_t groups = (M / (16 * (size_t)MT)) * (size_t)(Co / 16);
    return (int)((groups + 7) / 8);             // 8 waves (256 thr) per block
}

extern "C" void kernel_launch(void* const* d_in, const int* in_sizes, int n_in,
                              void* d_out, int out_size, void* d_ws, size_t ws_size,
                              hipStream_t stream)
{
    (void)in_sizes; (void)n_in; (void)out_size; (void)ws_size;

    const float* x       = (const float*)d_in[0];
    const int*   mask    = (const int*)  d_in[1];
    const int*   t       = (const int*)  d_in[2];
    const float* adj     = (const float*)d_in[3];
    const float* temb_w0 = (const float*)d_in[4];
    const float* temb_b0 = (const float*)d_in[5];
    const float* temb_w1 = (const float*)d_in[6];
    const float* temb_b1 = (const float*)d_in[7];
    const float* w_in    = (const float*)d_in[8];
    const float* b_in    = (const float*)d_in[9];
    const float* w_out   = (const float*)d_in[10];
    const float* b_out   = (const float*)d_in[11];
    const float* attn_w  = (const float*)d_in[12];
    const float* attn_b  = (const float*)d_in[13];
    const float* ln_a    = (const float*)d_in[14];
    const float* ln_b    = (const float*)d_in[15];
    const float* a_hat   = (const float*)d_in[16];
    const float* gcn_w1  = (const float*)d_in[17];
    const float* gcn_b1  = (const float*)d_in[18];
    const float* gcn_w2  = (const float*)d_in[19];
    const float* gcn_b2  = (const float*)d_in[20];
    const float* res_w1  = (const float*)d_in[21];
    const float* res_b1  = (const float*)d_in[22];
    const float* res_w2  = (const float*)d_in[23];
    const float* res_b2  = (const float*)d_in[24];
    const float* tp_w    = (const float*)d_in[25];
    const float* tp_b    = (const float*)d_in[26];

    const size_t BN = (size_t)B_SZ * N_SZ;      // 139264 (multiple of 64)

    float* ws  = (float*)d_ws;
    float* OUT = ws;  ws += BN * D_SZ;          // running residual stream
    float* TMP = ws;  ws += BN * D_SZ;          // LN out / attn out / h
    float* SCR = ws;  ws += BN * 288;           // Q|K|V, also Z (<= BN*256)
    float* Gh  = ws;  ws += BN * GH_SZ;         // GCN hidden (128)
    float* ST  = ws;  ws += (size_t)L_SZ * B_SZ * D_SZ;   // per-layer temb proj
    float* EMB = ws;  ws += (size_t)B_SZ * D_SZ;
    float* SW0 = ws;  ws += (size_t)B_SZ * EMD_SZ;        // swish(emb@w0+b0)
    float* SW1 = ws;  ws += (size_t)B_SZ * EMD_SZ;        // swish(temb)
    float* XLb = ws;  ws += BN * 5;
    float* LAPS= ws;  ws += 6 * N_SZ * N_SZ;

    // ---- Laplacians (adj + 5 a_hat) --------------------------------------
    lap_kernel<<<6, N_SZ * N_SZ, 0, stream>>>(adj, a_hat, LAPS);
    const float* LAP = LAPS;

    // ---- timestep embedding MLP + per-layer projections ------------------
    {
        int tot = B_SZ * D_SZ;
        temb_emb_kernel<<<(tot + 255) / 256, 256, 0, stream>>>(t, EMB, tot);
        gemm_wmma<96, 384, 4, 2, false, false>
            <<<gemm_blocks(B_SZ, EMD_SZ, 4), 256, 0, stream>>>(
            EMB, temb_w0, temb_b0, nullptr, nullptr, SW0, B_SZ, 1);
        gemm_wmma<384, 384, 4, 2, false, false>
            <<<gemm_blocks(B_SZ, EMD_SZ, 4), 256, 0, stream>>>(
            SW0, temb_w1, temb_b1, nullptr, nullptr, SW1, B_SZ, 1);
        for (int l = 0; l < L_SZ; ++l)
            gemm_wmma<384, 96, 4, 0, false, false>
                <<<gemm_blocks(B_SZ, D_SZ, 4), 256, 0, stream>>>(
                SW1, tp_w + (size_t)l * EMD_SZ * D_SZ, tp_b + l * D_SZ,
                nullptr, nullptr, ST + (size_t)l * B_SZ * D_SZ, B_SZ, 1);
    }

    // ---- input ChebConv: x[B,N,5] -> OUT[B,N,96] -------------------------
    {
        int tot5 = (int)(BN * 5);
        xl_kernel<<<(tot5 + 255) / 256, 256, 0, stream>>>(x, LAP, XLb, tot5);
        int totd = (int)(BN * D_SZ);
        incheb_kernel<<<(totd + 255) / 256, 256, 0, stream>>>(x, XLb, w_in, b_in, OUT, totd);
    }

    const int totD   = (int)(BN * D_SZ);
    const int totG   = (int)(BN * GH_SZ);
    const int lnBlks = (int)((BN + 7) / 8);
    const int gbD    = gemm_blocks(BN, D_SZ, 4);
    const int gbG    = gemm_blocks(BN, GH_SZ, 4);

    for (int i = 0; i < L_SZ; ++i) {
        const float* LH = LAPS + (size_t)(1 + i) * N_SZ * N_SZ;
        const float* aw = attn_w + (size_t)i * 4 * D_SZ * D_SZ;
        const float* ab = attn_b + (size_t)i * 4 * D_SZ;
        const float* la = ln_a + (size_t)i * 2 * D_SZ;
        const float* lb = ln_b + (size_t)i * 2 * D_SZ;
        float* Q  = SCR;
        float* Kb = SCR + BN * D_SZ;
        float* V  = SCR + 2 * BN * D_SZ;

        // --- self-attention sublayer (pre-LN) -----------------------------
        lnorm_kernel<<<lnBlks, 256, 0, stream>>>(OUT, la, lb, TMP, (int)BN);
        gemm_wmma<96, 96, 4, 0, false, false><<<gbD, 256, 0, stream>>>(
            TMP, aw, ab, nullptr, nullptr, Q, (int)BN, N_SZ);
        gemm_wmma<96, 96, 4, 0, false, false><<<gbD, 256, 0, stream>>>(
            TMP, aw + D_SZ * D_SZ, ab + D_SZ, nullptr, nullptr, Kb, (int)BN, N_SZ);
        gemm_wmma<96, 96, 4, 0, false, false><<<gbD, 256, 0, stream>>>(
            TMP, aw + 2 * D_SZ * D_SZ, ab + 2 * D_SZ, nullptr, nullptr, V, (int)BN, N_SZ);
        {
            int tot = B_SZ * H_SZ * N_SZ;
            attn_kernel<<<(tot + 255) / 256, 256, 0, stream>>>(Q, Kb, V, mask, TMP, tot);
        }
        gemm_wmma<96, 96, 4, 0, true, false><<<gbD, 256, 0, stream>>>(
            TMP, aw + 3 * D_SZ * D_SZ, ab + 3 * D_SZ, OUT, nullptr, OUT, (int)BN, N_SZ);

        // --- GraphNet sublayer (pre-LN, learnable Laplacian LH) -----------
        lnorm_kernel<<<lnBlks, 256, 0, stream>>>(OUT, la + D_SZ, lb + D_SZ, TMP, (int)BN);
        graphmix_kernel<<<(totD + 255) / 256, 256, 0, stream>>>(TMP, LH, SCR, D_SZ, totD);
        gemm_wmma<192, 128, 4, 1, false, false><<<gbG, 256, 0, stream>>>(
            SCR, gcn_w1 + (size_t)i * 2 * D_SZ * GH_SZ, gcn_b1 + i * GH_SZ,
            nullptr, nullptr, Gh, (int)BN, N_SZ);
        graphmix_kernel<<<(totG + 255) / 256, 256, 0, stream>>>(Gh, LH, SCR, GH_SZ, totG);
        gemm_wmma<256, 96, 4, 1, true, false><<<gbD, 256, 0, stream>>>(
            SCR, gcn_w2 + (size_t)i * 2 * GH_SZ * D_SZ, gcn_b2 + i * D_SZ,
            OUT, nullptr, OUT, (int)BN, N_SZ);

        // --- _ResChebGC_diff block (Laplacian LAP, + timestep embedding) --
        graphmix_kernel<<<(totD + 255) / 256, 256, 0, stream>>>(OUT, LAP, SCR, D_SZ, totD);
        gemm_wmma<192, 96, 4, 1, false, true><<<gbD, 256, 0, stream>>>(
            SCR, res_w1 + (size_t)i * 2 * D_SZ * D_SZ, res_b1 + i * D_SZ,
            nullptr, ST + (size_t)i * B_SZ * D_SZ, TMP, (int)BN, N_SZ);
        graphmix_kernel<<<(totD + 255) / 256, 256, 0, stream>>>(TMP, LAP, SCR, D_SZ, totD);
        gemm_wmma<192, 96, 4, 1, true, false><<<gbD, 256, 0, stream>>>(
            SCR, res_w2 + (size_t)i * 2 * D_SZ * D_SZ, res_b2 + i * D_SZ,
            OUT, nullptr, OUT, (int)BN, N_SZ);
    }

    // ---- output ChebConv: OUT -> [B,N,3] ---------------------------------
    graphmix_kernel<<<(totD + 255) / 256, 256, 0, stream>>>(OUT, LAP, SCR, D_SZ, totD);
    {
        int tot = (int)(BN * 3);
        outcheb_kernel<<<(tot + 255) / 256, 256, 0, stream>>>(SCR, w_out, b_out,
                                                              (float*)d_out, tot);
    }
}